// BachNetInferenceWithBeamSearch_55860344652404
// MI455X (gfx1250) — compile-verified
//
#include <hip/hip_runtime.h>
#include <hip/hip_bf16.h>
#include <math.h>

// ---------------- problem constants ----------------
#define PP     62          // P
#define DD     10112       // D = (5*32+1)*62 + (2*32+1)*2
#define HH     2048        // H
#define KSPLIT 32          // K-split for big GEMVs (32*316 = 10112 exactly)
#define CHUNK  316
#define KS2    16          // K-split for mid GEMV (16*128 = 2048)
#define CH2    128
#define LKS    16          // K-split for bass logits

typedef float v2f __attribute__((ext_vector_type(2)));
typedef float v8f __attribute__((ext_vector_type(8)));

__device__ __forceinline__ float selu_f(float x) {
    const float a = 1.6732632423543772f;
    const float s = 1.0507009873554805f;
    return x > 0.0f ? s * x : s * a * expm1f(x);
}

// ---------- K1: 3 big GEMVs fused (bw1 / aw1[0:D] / tw1[0:D]), split-K partials ----------
__global__ void gemv3_partial_kernel(const float* __restrict__ x,
                                     const float* __restrict__ W0,
                                     const float* __restrict__ W1,
                                     const float* __restrict__ W2,
                                     float* __restrict__ partials) {
    __shared__ float xs[CHUNK];
    const int mat = blockIdx.z;
    const float* __restrict__ W = (mat == 0) ? W0 : (mat == 1) ? W1 : W2;
    const int j  = blockIdx.x * 256 + threadIdx.x;   // 0..2047
    const int k0 = blockIdx.y * CHUNK;
    for (int i = threadIdx.x; i < CHUNK; i += 256) xs[i] = x[k0 + i];
    __syncthreads();
    float acc = 0.0f;
    const float* __restrict__ Wp = W + (size_t)k0 * HH + j;
    for (int k = 0; k < CHUNK; ++k) acc += xs[k] * Wp[(size_t)k * HH];
    partials[((size_t)mat * KSPLIT + blockIdx.y) * HH + j] = acc;
}

__global__ void gemv3_reduce_kernel(const float* __restrict__ partials,
                                    const float* __restrict__ bb1,
                                    const float* __restrict__ ab1,
                                    const float* __restrict__ tb1,
                                    float* __restrict__ t1,
                                    float* __restrict__ g_a,
                                    float* __restrict__ g_t) {
    const int mat = blockIdx.z;
    const int j   = blockIdx.x * 256 + threadIdx.x;
    float s = 0.0f;
    for (int c = 0; c < KSPLIT; ++c)
        s += partials[((size_t)mat * KSPLIT + c) * HH + j];
    if (mat == 0)      t1[j]  = selu_f(s + bb1[j]);     // bass layer1 (with selu)
    else if (mat == 1) g_a[j] = s + ab1[j];             // alto shared pre-activation
    else               g_t[j] = s + tb1[j];             // tenor shared pre-activation
}

// ---------- K2: mid GEMV  t2 = selu(t1 @ bw2 + bb2) ----------
__global__ void gemv_mid_partial_kernel(const float* __restrict__ xin,
                                        const float* __restrict__ W,
                                        float* __restrict__ partials) {
    __shared__ float xs[CH2];
    const int j  = blockIdx.x * 256 + threadIdx.x;
    const int k0 = blockIdx.y * CH2;
    if (threadIdx.x < CH2) xs[threadIdx.x] = xin[k0 + threadIdx.x];
    __syncthreads();
    float acc = 0.0f;
    const float* __restrict__ Wp = W + (size_t)k0 * HH + j;
    for (int k = 0; k < CH2; ++k) acc += xs[k] * Wp[(size_t)k * HH];
    partials[(size_t)blockIdx.y * HH + j] = acc;
}

__global__ void gemv_mid_reduce_kernel(const float* __restrict__ partials,
                                       const float* __restrict__ b,
                                       float* __restrict__ outv) {
    const int j = blockIdx.x * 256 + threadIdx.x;
    float s = 0.0f;
    for (int c = 0; c < KS2; ++c) s += partials[(size_t)c * HH + j];
    outv[j] = selu_f(s + b[j]);
}

// ---------- K3: bass logits (t2 @ bw3), split-K ----------
__global__ void bass_logits_partial_kernel(const float* __restrict__ t2,
                                           const float* __restrict__ bw3,
                                           float* __restrict__ partial_l) {
    const int t  = threadIdx.x;               // 64 threads
    const int k0 = blockIdx.x * (HH / LKS);
    float acc = 0.0f;
    if (t < PP) {
        for (int k = 0; k < HH / LKS; ++k)
            acc += t2[k0 + k] * bw3[(size_t)(k0 + k) * PP + t];
    }
    partial_l[blockIdx.x * 64 + t] = acc;
}

// ---------- K4: bass softmax + full stable descending sort (62 elements) ----------
__global__ void bass_head_kernel(const float* __restrict__ partial_l,
                                 const float* __restrict__ bb3,
                                 float* __restrict__ probs1,
                                 int* __restrict__ pb1) {
    __shared__ float v[64];
    __shared__ int used[64];
    const int t = threadIdx.x;                // 64 threads
    float acc = 0.0f;
    if (t < PP) {
        acc = bb3[t];
        for (int c = 0; c < LKS; ++c) acc += partial_l[c * 64 + t];
    }
    v[t] = (t < PP) ? acc : -INFINITY;
    __syncthreads();
    if (t == 0) {
        float mx = v[0];
        for (int i = 1; i < PP; ++i) mx = v[i] > mx ? v[i] : mx;
        float s = 0.0f;
        for (int i = 0; i < PP; ++i) { v[i] = expf(v[i] - mx); s += v[i]; }
        for (int i = 0; i < PP; ++i) v[i] /= s;
        for (int i = 0; i < PP; ++i) used[i] = 0;
        for (int it = 0; it < PP; ++it) {       // stable selection sort (desc)
            float bv = -INFINITY; int bi = -1;
            for (int i = 0; i < PP; ++i)
                if (!used[i] && v[i] > bv) { bv = v[i]; bi = i; }
            used[bi] = 1;
            probs1[it] = bv;
            pb1[it] = bi;
        }
    }
}

// ---------- K5: build padded layer-1 activations  h[64][2048] ----------
__global__ void build_h_kernel(const float* __restrict__ g,
                               const float* __restrict__ W1,
                               const int* __restrict__ idx1, int base1,
                               const int* __restrict__ idx2, int base2, int use2,
                               float* __restrict__ hbuf) {
    const int j = blockIdx.x * 256 + threadIdx.x;
    const int r = blockIdx.y;                 // 0..63
    float val = 0.0f;
    if (r < PP) {
        val = g[j] + W1[(size_t)(base1 + idx1[r]) * HH + j];
        if (use2) val += W1[(size_t)(base2 + idx2[r]) * HH + j];
        val = selu_f(val);
    }
    hbuf[(size_t)r * HH + j] = val;           // rows 62,63 zero-padded
}

// ---------- K6: WMMA f32 GEMM  out = selu(A[64x2048] @ W[2048x2048] + b) ----------
#define KC  64
#define LDP 68      // padded LDS row stride: 68*4=272B (16B aligned rows, conflict-free b64 reads)

#if defined(__gfx1250__)
typedef int v4i __attribute__((vector_size(16)));
typedef __attribute__((address_space(1))) v4i* gas_v4i_p;   // global (__device__) 16B vector ptr
typedef __attribute__((address_space(3))) v4i* las_v4i_p;   // LDS (__shared__) 16B vector ptr
#endif

__global__ __launch_bounds__(128)
void wmma_gemm_kernel(const float* __restrict__ A,
                      const float* __restrict__ W,
                      const float* __restrict__ bias,
                      float* __restrict__ out) {
    __shared__ float at[64][LDP];             // row-major A chunk: at[m][kk]
    const int tid   = threadIdx.x;            // 0..127 (4 waves of 32)
    const int wave  = tid >> 5;
    const int lane  = tid & 31;
    const int nc    = blockIdx.x * 64 + wave * 16 + (lane & 15);
    const int khalf = lane >> 4;              // 0 or 1
    const int ml    = lane & 15;
    v8f acc[4] = {};
    for (int k0 = 0; k0 < HH; k0 += KC) {
        __syncthreads();
#if defined(__gfx1250__) && __has_builtin(__builtin_amdgcn_global_load_async_to_lds_b128) && __has_builtin(__builtin_amdgcn_s_wait_asynccnt)
        // CDNA5 async copy: per-lane global addr -> per-lane LDS addr, ASYNCcnt-tracked
        for (int u = tid; u < 64 * 16; u += 128) {
            const int r  = u >> 4;
            const int c4 = (u & 15) << 2;     // 4-float (16B) units
            __builtin_amdgcn_global_load_async_to_lds_b128(
                (gas_v4i_p)(uintptr_t)(A + (size_t)r * HH + k0 + c4),
                (las_v4i_p)(uint32_t)(uintptr_t)&at[r][c4],
                0, 0);
        }
        __builtin_amdgcn_s_wait_asynccnt(0);
#else
        // fallback: coalesced load + LDS store
        for (int e = tid; e < 64 * KC; e += 128) {
            const int r  = e >> 6;
            const int kk = e & 63;
            at[r][kk] = A[(size_t)r * HH + k0 + kk];
        }
#endif
        // speculative prefetch of next B chunk rows for this block's 64-column stripe
        if (k0 + KC < HH)
            __builtin_prefetch(W + (size_t)(k0 + KC + (tid & 63)) * HH + blockIdx.x * 64, 0, 0);
        __syncthreads();
        for (int k = 0; k < KC; k += 4) {
            const int kb = k0 + k + khalf * 2;
            v2f bf;
            bf.x = W[(size_t)kb * HH + nc];
            bf.y = W[(size_t)(kb + 1) * HH + nc];
            const int kl = k + khalf * 2;     // even -> 8B-aligned LDS b64 read
#pragma unroll
            for (int mt = 0; mt < 4; ++mt) {
                const v2f af = *(const v2f*)&at[mt * 16 + ml][kl];
                acc[mt] = __builtin_amdgcn_wmma_f32_16x16x4_f32(
                    false, af, false, bf, (short)0, acc[mt], false, false);
            }
        }
    }
    const float b = bias[nc];
#pragma unroll
    for (int mt = 0; mt < 4; ++mt) {
#pragma unroll
        for (int i = 0; i < 8; ++i) {
            const int m = mt * 16 + khalf * 8 + i;   // C layout: M = 8*(lane/16)+vgpr
            out[(size_t)m * HH + nc] = selu_f(acc[mt][i] + b);
        }
    }
}

// ---------- K7: per-row head  flat[r][j] = softmax(u[r] @ W3 + b3)[j] * probs_in[r] ----------
__global__ void head_kernel(const float* __restrict__ u,
                            const float* __restrict__ W3,
                            const float* __restrict__ b3,
                            const float* __restrict__ probs_in,
                            float* __restrict__ flat) {
    __shared__ float vals[64];
    __shared__ float mxs, sums;
    const int r = blockIdx.x;                 // 0..61
    const int j = threadIdx.x;                // 64 threads
    float acc = 0.0f;
    if (j < PP) {
        const float* __restrict__ ur = u + (size_t)r * HH;
        for (int k = 0; k < HH; ++k) acc += ur[k] * W3[(size_t)k * PP + j];
        acc += b3[j];
    }
    vals[j] = (j < PP) ? acc : -INFINITY;
    __syncthreads();
    if (j == 0) {
        float mx = vals[0];
        for (int i = 1; i < PP; ++i) mx = vals[i] > mx ? vals[i] : mx;
        float s = 0.0f;
        for (int i = 0; i < PP; ++i) s += expf(vals[i] - mx);
        mxs = mx; sums = s;
    }
    __syncthreads();
    if (j < PP)
        flat[r * PP + j] = expf(vals[j] - mxs) / sums * probs_in[r];
}

// ---------- K8: top-62 of 3844 (stable: ties -> smallest flat index) ----------
__global__ void top62_kernel(const float* __restrict__ flat,
                             const int* __restrict__ pb_in,
                             const int* __restrict__ pa_in,
                             float* __restrict__ probs_out,
                             int* __restrict__ pb_out,
                             int* __restrict__ pa_out,
                             float* __restrict__ final_out,
                             int mode) {
    __shared__ float vals[3968];
    __shared__ float rv[256];
    __shared__ int   ri[256];
    __shared__ float selp[PP];
    __shared__ int   seli[PP];
    const int t = threadIdx.x;                // 256 threads
    for (int i = t; i < 3968; i += 256)
        vals[i] = (i < PP * PP) ? flat[i] : -INFINITY;
    __syncthreads();
    for (int it = 0; it < PP; ++it) {
        float bv = -INFINITY; int bi = 0x7fffffff;
        for (int i = t; i < PP * PP; i += 256) {
            const float v = vals[i];
            if (v > bv || (v == bv && i < bi)) { bv = v; bi = i; }
        }
        rv[t] = bv; ri[t] = bi;
        __syncthreads();
        for (int s = 128; s > 0; s >>= 1) {
            if (t < s) {
                if (rv[t + s] > rv[t] || (rv[t + s] == rv[t] && ri[t + s] < ri[t])) {
                    rv[t] = rv[t + s]; ri[t] = ri[t + s];
                }
            }
            __syncthreads();
        }
        if (t == 0) { selp[it] = rv[0]; seli[it] = ri[0]; vals[ri[0]] = -INFINITY; }
        __syncthreads();
    }
    if (t < PP) {
        const int idx = seli[t];
        const int row = idx / PP;
        const int col = idx % PP;
        if (mode == 2) {
            probs_out[t] = selp[t];
            pb_out[t]    = pb_in[row];
            pa_out[t]    = col;
        } else {
            final_out[t * 4 + 0] = selp[t];
            final_out[t * 4 + 1] = (float)pb_in[row];
            final_out[t * 4 + 2] = (float)pa_in[row];
            final_out[t * 4 + 3] = (float)col;
        }
    }
}

// ---------------- host-side orchestration ----------------
extern "C" void kernel_launch(void* const* d_in, const int* in_sizes, int n_in,
                              void* d_out, int out_size, void* d_ws, size_t ws_size,
                              hipStream_t stream) {
    const float* x   = (const float*)d_in[0];
    const float* bw1 = (const float*)d_in[1];
    const float* bb1 = (const float*)d_in[2];
    const float* bw2 = (const float*)d_in[3];
    const float* bb2 = (const float*)d_in[4];
    const float* bw3 = (const float*)d_in[5];
    const float* bb3 = (const float*)d_in[6];
    const float* aw1 = (const float*)d_in[7];
    const float* ab1 = (const float*)d_in[8];
    const float* aw2 = (const float*)d_in[9];
    const float* ab2 = (const float*)d_in[10];
    const float* aw3 = (const float*)d_in[11];
    const float* ab3 = (const float*)d_in[12];
    const float* tw1 = (const float*)d_in[13];
    const float* tb1 = (const float*)d_in[14];
    const float* tw2 = (const float*)d_in[15];
    const float* tb2 = (const float*)d_in[16];
    const float* tw3 = (const float*)d_in[17];
    const float* tb3 = (const float*)d_in[18];
    float* out = (float*)d_out;

    // workspace layout (floats)
    float* ws = (float*)d_ws;
    float* partials  = ws;                      // 3*32*2048 = 196608 (reused for mid GEMV)
    float* t1        = ws + 196608;             // 2048
    float* t2        = t1 + 2048;               // 2048
    float* g_a       = t2 + 2048;               // 2048
    float* g_t       = g_a + 2048;              // 2048
    float* partial_l = g_t + 2048;              // 1024
    float* probs1    = partial_l + 1024;        // 64
    float* probs2    = probs1 + 64;             // 64
    int*   pb1       = (int*)(probs2 + 64);     // 64
    int*   pb2       = pb1 + 64;                // 64
    int*   pa2       = pb2 + 64;                // 64
    float* flat      = (float*)(pa2 + 64);      // 3904
    float* hbuf      = flat + 3904;             // 64*2048
    float* ubuf      = hbuf + 64 * 2048;        // 64*2048

    // Stage 0: the three 83MB GEMVs (bass layer1 + shared alto/tenor layer1)
    gemv3_partial_kernel<<<dim3(HH / 256, KSPLIT, 3), 256, 0, stream>>>(x, bw1, aw1, tw1, partials);
    gemv3_reduce_kernel<<<dim3(HH / 256, 1, 3), 256, 0, stream>>>(partials, bb1, ab1, tb1, t1, g_a, g_t);

    // Bass layers 2-3 + softmax + full sort
    gemv_mid_partial_kernel<<<dim3(HH / 256, KS2), 256, 0, stream>>>(t1, bw2, partials);
    gemv_mid_reduce_kernel<<<HH / 256, 256, 0, stream>>>(partials, bb2, t2);
    bass_logits_partial_kernel<<<LKS, 64, 0, stream>>>(t2, bw3, partial_l);
    bass_head_kernel<<<1, 64, 0, stream>>>(partial_l, bb3, probs1, pb1);

    // Alto: layer1 via shared GEMV + one-hot row gather, then WMMA GEMM, head, top-62
    build_h_kernel<<<dim3(HH / 256, 64), 256, 0, stream>>>(g_a, aw1, pb1, DD, pb1, 0, 0, hbuf);
    wmma_gemm_kernel<<<HH / 64, 128, 0, stream>>>(hbuf, aw2, ab2, ubuf);
    head_kernel<<<PP, 64, 0, stream>>>(ubuf, aw3, ab3, probs1, flat);
    top62_kernel<<<1, 256, 0, stream>>>(flat, pb1, pb1, probs2, pb2, pa2, out, 2);

    // Tenor: layer1 gathers two one-hot rows, WMMA GEMM, head, final top-62 -> out
    build_h_kernel<<<dim3(HH / 256, 64), 256, 0, stream>>>(g_t, tw1, pb2, DD, pa2, DD + PP, 1, hbuf);
    wmma_gemm_kernel<<<HH / 64, 128, 0, stream>>>(hbuf, tw2, tb2, ubuf);
    head_kernel<<<PP, 64, 0, stream>>>(ubuf, tw3, tb3, probs2, flat);
    top62_kernel<<<1, 256, 0, stream>>>(flat, pb2, pa2, probs2, pb2, pa2, out, 3);
}